// GraphSAGE_14190571946097
// MI455X (gfx1250) — compile-verified
//
#include <hip/hip_runtime.h>

typedef __attribute__((ext_vector_type(2))) float v2f;
typedef __attribute__((ext_vector_type(8))) float v8f;

#define N_NODES 100000
#define N_EDGES 1600000
#define N_PAIRS 500000
#define D 64

static __device__ __forceinline__ v8f wmma_f32_k4(v2f a, v2f b, v8f c) {
  return __builtin_amdgcn_wmma_f32_16x16x4_f32(false, a, false, b, (short)0, c,
                                               false, false);
}

// ---------------------------------------------------------------- utilities

__global__ void zero_f32(float* __restrict__ p, long n) {
  long i = (long)blockIdx.x * blockDim.x + threadIdx.x;
  long stride = (long)gridDim.x * blockDim.x;
  for (; i < n; i += stride) p[i] = 0.0f;
}

__global__ void degree_kernel(const int* __restrict__ dst, float* __restrict__ deg, int nE) {
  int i = blockIdx.x * blockDim.x + threadIdx.x;
  int stride = gridDim.x * blockDim.x;
  for (; i < nE; i += stride) atomicAdd(&deg[dst[i]], 1.0f);
}

// one thread per (edge, dim): lanes cover consecutive dims of one source row
// -> coalesced 128B row reads, consecutive atomic adds into the dst row (L2-resident).
__global__ void scatter_kernel(const float* __restrict__ feat,
                               const int* __restrict__ src,
                               const int* __restrict__ dst,
                               float* __restrict__ neigh, int nE) {
  long total = (long)nE * D;
  long i = (long)blockIdx.x * blockDim.x + threadIdx.x;
  long stride = (long)gridDim.x * blockDim.x;
  for (; i < total; i += stride) {
    int e = (int)(i >> 6);
    int d = (int)(i & 63);
    atomicAdd(&neigh[(long)dst[e] * D + d], feat[(long)src[e] * D + d]);
  }
}

// ------------------------------------------------- fused SAGE layer (WMMA f32)
// Y[i,:] = relu( X[i,:] @ Ws + (NB[i,:]/max(deg[i],1)) @ Wn + bias )
// Weights staged in LDS (pair-interleaved so each B fragment is one ds_load_b64);
// A row preloaded into registers; K-loop is back-to-back v_wmma_f32_16x16x4_f32.
__global__ __launch_bounds__(256) void sage_gemm(
    const float* __restrict__ X, const float* __restrict__ NB,
    const float* __restrict__ deg,
    const float* __restrict__ Ws, const float* __restrict__ Wn,
    const float* __restrict__ bias,
    float* __restrict__ Y, int nRows) {
  // lds layout per matrix: [(k>>1)*128 + c*2 + (k&1)], k<64, c<64 -> 4096 floats
  __shared__ float ldsW[2][64 * 64];

  for (int i = threadIdx.x; i < 64 * 64; i += 256) {
    int k = i >> 6, c = i & 63;
    int a = (k >> 1) * 128 + c * 2 + (k & 1);
    ldsW[0][a] = Ws[i];
    ldsW[1][a] = Wn[i];
  }
  __syncthreads();

  const int lane = threadIdx.x & 31;
  const int wave = threadIdx.x >> 5;
  const int rowBase = (blockIdx.x * 8 + wave) * 16;
  const int r  = lane & 15;   // A: row in tile / B,D: column in tile
  const int kh = lane >> 4;   // K-pair selector (lanes 0-15: K{0,1}, 16-31: K{2,3})

  int row  = rowBase + r;
  int rowc = row < nRows ? row : nRows - 1;   // clamp: EXEC stays all-ones
  float invd = __builtin_amdgcn_rcpf(fmaxf(deg[rowc], 1.0f));

  // preload A fragments: 16 K-steps for self + neighbor halves
  v2f aS[16], aN[16];
#pragma unroll
  for (int q = 0; q < 16; ++q) {
    int ka = q * 4 + kh * 2;
    aS[q] = *(const v2f*)(X + (long)rowc * D + ka);
    v2f n = *(const v2f*)(NB + (long)rowc * D + ka);
    aN[q].x = n.x * invd;
    aN[q].y = n.y * invd;
  }

  v8f acc[4] = {};
#pragma unroll
  for (int t = 0; t < 4; ++t) {
    const int col2 = (t * 16 + r) * 2;
    v2f bF[16];
#pragma unroll
    for (int q = 0; q < 16; ++q)
      bF[q] = *(const v2f*)&ldsW[0][q * 128 + col2];
#pragma unroll
    for (int q = 0; q < 16; ++q)
      acc[t] = wmma_f32_k4(aS[q], bF[q], acc[t]);
#pragma unroll
    for (int q = 0; q < 16; ++q)
      bF[q] = *(const v2f*)&ldsW[1][q * 128 + col2];
#pragma unroll
    for (int q = 0; q < 16; ++q)
      acc[t] = wmma_f32_k4(aN[q], bF[q], acc[t]);
  }

  // epilogue: bias + relu + store (D layout: VGPR v -> row v/v+8, lane -> col)
#pragma unroll
  for (int t = 0; t < 4; ++t) {
    int col = t * 16 + r;
    float bv = bias[col];
#pragma unroll
    for (int v = 0; v < 8; ++v) {
      int orow = rowBase + v + kh * 8;
      if (orow < nRows) {
        float val = acc[t][v] + bv;
        Y[(long)orow * D + col] = val > 0.0f ? val : 0.0f;
      }
    }
  }
}

// ------------------------------------------------------------- pair MLP head
// e = [h[x1] | h[x2] | |h[x1]-h[x2]|]  (K=192)
// hid = relu(e @ W1 + b1) via WMMA (W1 in LDS); out = hid @ W2 + b2 via LDS dot.
__global__ __launch_bounds__(256) void head_kernel(
    const float* __restrict__ H,
    const int* __restrict__ x1, const int* __restrict__ x2,
    const float* __restrict__ W1, const float* __restrict__ b1,
    const float* __restrict__ W2, const float* __restrict__ b2,
    float* __restrict__ out, int nPairs) {
  // 48 KB: W1 pair-interleaved; first 8K floats aliased for hid tiles afterwards
  __shared__ float ldsW1[96 * 128];

  for (int i = threadIdx.x; i < 192 * 64; i += 256) {
    int k = i >> 6, c = i & 63;
    ldsW1[(k >> 1) * 128 + c * 2 + (k & 1)] = W1[i];
  }
  __syncthreads();

  const int lane = threadIdx.x & 31;
  const int wave = threadIdx.x >> 5;
  const int pairBase = (blockIdx.x * 8 + wave) * 16;
  const int r  = lane & 15;
  const int kh = lane >> 4;

  int p  = pairBase + r;
  int pc = p < nPairs ? p : nPairs - 1;
  const float* A1 = H + (long)x1[pc] * D;
  const float* A2 = H + (long)x2[pc] * D;

  v8f acc[4] = {};
  for (int k = 0; k < 192; k += 4) {
    int ka = k + kh * 2;
    v2f a;
    if (ka < 64) {
      a = *(const v2f*)(A1 + ka);
    } else if (ka < 128) {
      a = *(const v2f*)(A2 + ka - 64);
    } else {
      v2f u = *(const v2f*)(A1 + ka - 128);
      v2f w = *(const v2f*)(A2 + ka - 128);
      a.x = fabsf(u.x - w.x);
      a.y = fabsf(u.y - w.y);
    }
#pragma unroll
    for (int t = 0; t < 4; ++t) {
      v2f b = *(const v2f*)&ldsW1[(ka >> 1) * 128 + (t * 16 + r) * 2];
      acc[t] = wmma_f32_k4(a, b, acc[t]);
    }
  }
  __syncthreads();   // all waves done reading W1 before aliasing its LDS

  // relu(acc + b1) -> per-wave 16x64 tile in (aliased) LDS
  float* myHid = &ldsW1[wave * 16 * D];
#pragma unroll
  for (int t = 0; t < 4; ++t) {
    int col = t * 16 + r;
    float bv = b1[col];
#pragma unroll
    for (int v = 0; v < 8; ++v) {
      float val = acc[t][v] + bv;
      myHid[(v + kh * 8) * D + col] = val > 0.0f ? val : 0.0f;
    }
  }
  __syncthreads();   // order cross-lane LDS stores vs loads (uniform)

  if (lane < 16) {
    int p2 = pairBase + lane;
    if (p2 < nPairs) {
      float acc0 = b2[0], acc1 = b2[1];
#pragma unroll 8
      for (int c = 0; c < D; ++c) {
        float v = myHid[lane * D + c];
        acc0 += v * W2[c * 2 + 0];
        acc1 += v * W2[c * 2 + 1];
      }
      out[(long)p2 * 2 + 0] = acc0;
      out[(long)p2 * 2 + 1] = acc1;
    }
  }
}

// ------------------------------------------------------------------- launch

extern "C" void kernel_launch(void* const* d_in, const int* in_sizes, int n_in,
                              void* d_out, int out_size, void* d_ws, size_t ws_size,
                              hipStream_t stream) {
  const float* h   = (const float*)d_in[0];
  const int*   src = (const int*)d_in[1];
  const int*   dst = (const int*)d_in[2];
  const int*   x1  = (const int*)d_in[3];
  const int*   x2  = (const int*)d_in[4];
  const float* Ws0 = (const float*)d_in[5];
  const float* Wn0 = (const float*)d_in[6];
  const float* b0  = (const float*)d_in[7];
  const float* Ws1 = (const float*)d_in[8];
  const float* Wn1 = (const float*)d_in[9];
  const float* b1  = (const float*)d_in[10];
  const float* W1  = (const float*)d_in[11];
  const float* bl1 = (const float*)d_in[12];
  const float* W2  = (const float*)d_in[13];
  const float* bl2 = (const float*)d_in[14];
  float* out = (float*)d_out;

  // workspace layout (floats): deg[N] | neigh[N*64] | t1[N*64] | t2[N*64]  (~77 MB)
  float* deg   = (float*)d_ws;
  float* neigh = deg + N_NODES;
  float* t1    = neigh + (size_t)N_NODES * D;
  float* t2    = t1 + (size_t)N_NODES * D;

  const int sageBlocks = (N_NODES + 127) / 128;   // 128 rows / block (8 waves x 16)
  const int headBlocks = (N_PAIRS + 127) / 128;

  zero_f32<<<2048, 256, 0, stream>>>(deg, (long)N_NODES * (1 + D));  // deg+neigh contiguous
  degree_kernel<<<1024, 256, 0, stream>>>(dst, deg, N_EDGES);
  scatter_kernel<<<4096, 256, 0, stream>>>(h, src, dst, neigh, N_EDGES);
  sage_gemm<<<sageBlocks, 256, 0, stream>>>(h, neigh, deg, Ws0, Wn0, b0, t1, N_NODES);

  zero_f32<<<2048, 256, 0, stream>>>(neigh, (long)N_NODES * D);
  scatter_kernel<<<4096, 256, 0, stream>>>(t1, src, dst, neigh, N_EDGES);
  sage_gemm<<<sageBlocks, 256, 0, stream>>>(t1, neigh, deg, Ws1, Wn1, b1, t2, N_NODES);

  head_kernel<<<headBlocks, 256, 0, stream>>>(t2, x1, x2, W1, bl1, W2, bl2, out, N_PAIRS);
}